// FDLT_47545287967040
// MI455X (gfx1250) — compile-verified
//
#include <hip/hip_runtime.h>
#include <hip/hip_bf16.h>
#include <stdint.h>

// ---------------------------------------------------------------------------
// FDLT (B=64, BATCH=512) on gfx1250.
//
// Fused formulation:
//   out[b,m,l] = sum_j psiHat[b,m,j] * E_m[j,l]
//   E_m[j,l]   = Cm[m] * sum_n XF_{parity}[j,n] * W_m[l,n]
// W_m = dense-ified [64 x 128] sparse Wigner block for order m.
// E is stored TRANSPOSED as Et[m][l][j] so the main GEMM's B-operand lane
// fragments are contiguous 8B pairs in LDS (single ds_load_b64 per operand).
//
// Matmuls: V_WMMA_F32_16X16X4_F32 (full-fp32 WMMA, wave32).
// Data movement: GLOBAL_LOAD_ASYNC_TO_LDS_B128 (ASYNCcnt) for LDS staging.
// ---------------------------------------------------------------------------

typedef __attribute__((ext_vector_type(2))) float v2f;
typedef __attribute__((ext_vector_type(8))) float v8f;

#define BB   64      // bandlimit B
#define MM   127     // 2B-1
#define NN   128     // 2B
#define NB   512     // batch

// Async global -> LDS copy of 16 bytes per lane (GV mode, 64-bit vaddr).
__device__ __forceinline__ void async_copy_b128(const void* gptr, void* lptr) {
    uint32_t lds  = (uint32_t)(uintptr_t)lptr;   // low 32 bits == LDS byte offset
    uint64_t gadr = (uint64_t)(uintptr_t)gptr;
    asm volatile("global_load_async_to_lds_b128 %0, %1, off"
                 :: "v"(lds), "v"(gadr) : "memory");
}
__device__ __forceinline__ void async_wait0() {
    asm volatile("s_wait_asynccnt 0x0" ::: "memory");
}

// ---------------------------------------------------------------------------
// 1) zero the dense W workspace
// ---------------------------------------------------------------------------
__global__ void fdlt_zero(float* __restrict__ p, int n) {
    int i = blockIdx.x * blockDim.x + threadIdx.x;
    if (i < n) p[i] = 0.0f;
}

// ---------------------------------------------------------------------------
// 2) scatter COO (D_val, D_row, D_col) into dense W[127][64][128]
//    row = l + m*64, col = n + m*128  (entries unique -> plain stores)
// ---------------------------------------------------------------------------
__global__ void fdlt_scatter(const float* __restrict__ val,
                             const int* __restrict__ row,
                             const int* __restrict__ col,
                             int nnz, float* __restrict__ W) {
    int i = blockIdx.x * blockDim.x + threadIdx.x;
    if (i >= nnz) return;
    int r  = row[i];
    int mi = r >> 6;        // r / 64
    int l  = r & 63;        // r % 64
    int n  = col[i] & 127;  // col % 128
    W[((size_t)mi * BB + l) * NN + n] = val[i];
}

// ---------------------------------------------------------------------------
// 3) build Et[m][l][j] = Cm[m] * sum_n XF[j,n] * W_m[l,n]
//    grid: (127 m, 8 j-tiles), 128 threads = 4 waves, wave w -> l-tile w.
//    WMMA f32 16x16x4: A[jrow][ksub]=XF[j, n0+ksub], B[ksub][lcol]=Wm[l, n0+ksub]
// ---------------------------------------------------------------------------
__global__ __launch_bounds__(128) void fdlt_build_E(
    const float* __restrict__ XFc, const float* __restrict__ XFs,
    const float* __restrict__ Cm,  const float* __restrict__ W,
    float* __restrict__ Et)
{
    int m     = blockIdx.x;
    int ktile = blockIdx.y;            // j-tile, 0..7
    int tid   = threadIdx.x;
    int lane  = tid & 31;
    int wave  = tid >> 5;              // l-tile, 0..3

    // B-1 = 63 odd: cInd = odd m-index -> XFc, sInd = even m-index -> XFs
    const float* XF = (m & 1) ? XFc : XFs;
    const float* Wm = W + (size_t)m * BB * NN;

    int row = lane & 15;
    int kp  = (lane >> 4) << 1;        // 0 for lanes 0-15, 2 for lanes 16-31

    const float* Arow = XF + (size_t)(ktile * 16 + row) * NN;   // XF[j,*]
    const float* Brow = Wm + (size_t)(wave  * 16 + row) * NN;   // W_m[l,*]

    v8f acc = {};
#pragma unroll
    for (int n0 = 0; n0 < NN; n0 += 4) {
        v2f a = *(const v2f*)(Arow + n0 + kp);
        v2f b = *(const v2f*)(Brow + n0 + kp);
        acc = __builtin_amdgcn_wmma_f32_16x16x4_f32(
            /*neg_a=*/false, a, /*neg_b=*/false, b,
            /*c_mod=*/(short)0, acc, /*reuse_a=*/false, /*reuse_b=*/false);
    }

    float cm   = Cm[m];
    int   l    = wave * 16 + (lane & 15);
    int   jbas = ktile * 16 + ((lane < 16) ? 0 : 8);
    // transposed store: Et[m][l][j]
#pragma unroll
    for (int v = 0; v < 8; ++v)
        Et[(size_t)m * BB * NN + (size_t)l * NN + (jbas + v)] = cm * acc[v];
}

// ---------------------------------------------------------------------------
// 4) main batched GEMM: out[b,m,l] = sum_j psiHat[b,m,j] * Et_m[l,j]
//    grid: (127 m, 16 batch-tiles of 32), 256 threads = 8 waves.
//    LDS: Et_m (64x128 = 32KB) + A tile (32x128 = 16KB), filled with
//    GLOBAL_LOAD_ASYNC_TO_LDS_B128.  Each wave: one 16x16 C tile,
//    K=128 -> 32 chained v_wmma_f32_16x16x4_f32, operands via ds_load_b64.
// ---------------------------------------------------------------------------
__global__ __launch_bounds__(256) void fdlt_main(
    const float* __restrict__ psiHat, const float* __restrict__ Et,
    float* __restrict__ out)
{
    __shared__ float sB[BB * NN];   // Et_m[l][j]  (32KB)
    __shared__ float sA[32 * NN];   // A[b'][j]    (16KB)

    int m   = blockIdx.x;
    int b0  = blockIdx.y * 32;
    int tid = threadIdx.x;

    // async-stage Et_m -> LDS (2048 x b128, 8 per thread)
    {
        const float4* Bg = (const float4*)(Et + (size_t)m * BB * NN);
        float4*       sB4 = (float4*)sB;
        for (int i = tid; i < (BB * NN) / 4; i += 256)
            async_copy_b128(Bg + i, sB4 + i);
    }
    // async-stage A tile -> LDS: rows b0..b0+31 of psiHat[:, m, :] (1024 x b128)
    {
        float4* sA4 = (float4*)sA;
        for (int i = tid; i < 32 * (NN / 4); i += 256) {
            int r  = i >> 5;        // 0..31  (row)
            int cc = i & 31;        // 0..31  (float4 within row)
            async_copy_b128((const float4*)(psiHat + (size_t)(b0 + r) * (MM * NN)
                                                   + (size_t)m * NN) + cc,
                            sA4 + (r << 5) + cc);
        }
    }
    async_wait0();
    __syncthreads();

    int lane  = tid & 31;
    int wave  = tid >> 5;
    int btile = wave >> 2;          // 0..1
    int ltile = wave & 3;           // 0..3
    int row   = lane & 15;
    int kp    = (lane >> 4) << 1;   // 0 or 2

    const float* Arow = sA + (size_t)(btile * 16 + row) * NN; // A[row][*]
    const float* Brow = sB + (size_t)(ltile * 16 + row) * NN; // B[*][col] = Et[l][*]

    v8f acc = {};
#pragma unroll
    for (int j0 = 0; j0 < NN; j0 += 4) {
        v2f a = *(const v2f*)(Arow + j0 + kp);
        v2f b = *(const v2f*)(Brow + j0 + kp);
        acc = __builtin_amdgcn_wmma_f32_16x16x4_f32(
            /*neg_a=*/false, a, /*neg_b=*/false, b,
            /*c_mod=*/(short)0, acc, /*reuse_a=*/false, /*reuse_b=*/false);
    }

    // store C: VGPR v -> row (v or v+8), col = lane&15
    int col   = ltile * 16 + (lane & 15);
    int rbase = btile * 16 + ((lane < 16) ? 0 : 8);
#pragma unroll
    for (int v = 0; v < 8; ++v) {
        int bb = b0 + rbase + v;
        out[(size_t)bb * (MM * BB) + (size_t)m * BB + col] = acc[v];
    }
}

// ---------------------------------------------------------------------------
// launch
// ---------------------------------------------------------------------------
extern "C" void kernel_launch(void* const* d_in, const int* in_sizes, int n_in,
                              void* d_out, int out_size, void* d_ws, size_t ws_size,
                              hipStream_t stream) {
    const float* psiHat = (const float*)d_in[0];   // [512,127,128]
    const float* Cm     = (const float*)d_in[1];   // [127]
    const float* XFc    = (const float*)d_in[2];   // [128,128]
    const float* XFs    = (const float*)d_in[3];   // [128,128]
    const float* D_val  = (const float*)d_in[4];
    const int*   D_row  = (const int*)d_in[5];
    const int*   D_col  = (const int*)d_in[6];
    int nnz = in_sizes[4];

    float* W  = (float*)d_ws;                      // [127][64][128]
    float* Et = W + (size_t)MM * BB * NN;          // [127][64][128] (l-major)
    float* out = (float*)d_out;                    // [512,127,64]

    int wn = MM * BB * NN;
    fdlt_zero<<<(wn + 255) / 256, 256, 0, stream>>>(W, wn);
    fdlt_scatter<<<(nnz + 255) / 256, 256, 0, stream>>>(D_val, D_row, D_col, nnz, W);

    dim3 ge(MM, NN / 16);                          // (127, 8)
    fdlt_build_E<<<ge, 128, 0, stream>>>(XFc, XFs, Cm, W, Et);

    dim3 gm(MM, NB / 32);                          // (127, 16)
    fdlt_main<<<gm, 256, 0, stream>>>(psiHat, Et, out);
}